// FFM_21912923144763
// MI455X (gfx1250) — compile-verified
//
#include <hip/hip_runtime.h>
#include <math.h>

typedef __attribute__((ext_vector_type(2))) float v2f;
typedef __attribute__((ext_vector_type(8))) float v8f;

#define FEAT    221
#define FEATP   224      // padded to 14*16
#define NTILE   14       // FEATP/16 column tiles
#define KTILES  56       // FEATP/4 K steps
#define ASTRIDE 228      // LDS row stride (floats): 16B-aligned, conflict-free rows
#define TSTRIDE 20       // LDS T-scratch row stride: conflict-free rows
#define WAVE_LDS (ASTRIDE * 16 + TSTRIDE * 16)

// ---------------------------------------------------------------------------
// Kernel 1: build M (224x224, strict upper triangle, zero-padded) directly in
// the swizzled per-lane layout of the V_WMMA_F32_16X16X4_F32 B-operand:
//   Mw[((n*56 + kt)*32 + lane)*2 + v]  <-  M[4*kt + v + 2*(lane>>4)][16*n + (lane&15)]
// so the hot loop reads one coalesced b64 per lane per WMMA.
// ---------------------------------------------------------------------------
__global__ __launch_bounds__(256) void ffm_build_M(const float* __restrict__ fm,
                                                   float* __restrict__ Mw) {
  int tid = blockIdx.x * blockDim.x + threadIdx.x;
  if (tid >= FEATP * FEATP) return;
  int i = tid / FEATP;  // K index (row of M)
  int j = tid % FEATP;  // column of M
  float val = 0.f;
  if (i < j && j < FEAT) {
    int fi = (i < 13) ? i : 13 + ((i - 13) >> 3);
    int fj = (j < 13) ? j : 13 + ((j - 13) >> 3);
    const float* gi = fm + (i * 39 + fj) * 8;  // field_matrix[i, field(j), :]
    const float* gj = fm + (j * 39 + fi) * 8;  // field_matrix[j, field(i), :]
#pragma unroll
    for (int k = 0; k < 8; ++k) val += gi[k] * gj[k];
  }
  int n    = j >> 4;
  int cl   = j & 15;
  int kt   = i >> 2;
  int rem  = i & 3;
  int v    = rem & 1;
  int hi   = rem >> 1;
  int lane = cl + 16 * hi;
  Mw[((n * KTILES + kt) * 32 + lane) * 2 + v] = val;
}

// ---------------------------------------------------------------------------
// Kernel 2: per wave, 16 samples. Gather A tile (16x224) to LDS, then
//   T = A*M via f32 WMMA (56 K-steps per 16-col tile),
//   interact = diag(T * A^T) accumulated in one 16x16 WMMA accumulator,
//   out = sigmoid(a.W + b_lin + interact).
// ---------------------------------------------------------------------------
__global__ __launch_bounds__(128) void ffm_main(
    const float* __restrict__ x_dense, const int* __restrict__ x_sparse,
    const float* __restrict__ emb, const float* __restrict__ W_lin,
    const float* __restrict__ b_lin, const float* __restrict__ Mw,
    float* __restrict__ out, int nB) {
  __shared__ float smem[4 * WAVE_LDS];  // 63488 bytes
  const int wave = threadIdx.x >> 5;
  const int lane = threadIdx.x & 31;
  const int r    = lane & 15;   // sample row within tile / matrix row
  const int hi   = lane >> 4;   // lane half
  float* As = smem + wave * WAVE_LDS;
  float* Ts = As + ASTRIDE * 16;
  const int base = blockIdx.x * 64 + wave * 16;

  // ---- gather A tile: [dense(13) | 26 embeddings x 8 | zero pad(3)] ----
  for (int rr = 0; rr < 16; ++rr) {
    int b = base + rr;
    for (int j = lane; j < FEATP; j += 32) {
      float v = 0.f;
      if (b < nB) {
        if (j < 13) {
          v = x_dense[b * 13 + j];
        } else if (j < FEAT) {
          int s = (j - 13) >> 3, k = (j - 13) & 7;
          int idx = x_sparse[b * 26 + s];
          v = emb[(s * 1000 + idx) * 8 + k];
        }
      }
      As[rr * ASTRIDE + j] = v;
    }
  }
  __syncthreads();

  // ---- linear term: 2 lanes per row, halves reduced via shfl ----
  float part = 0.f;
  {
    int j0 = hi * 112;
    int j1 = hi ? FEAT : 112;
    for (int j = j0; j < j1; ++j) part += As[r * ASTRIDE + j] * W_lin[j];
  }
  part += __shfl_down(part, 16);
  float lin = part + b_lin[0];  // valid in lanes 0..15

  // ---- quadratic form via fp32 WMMA ----
  v8f acc2 = {};  // accumulates T * A^T (16x16); diagonal = interact
  for (int n = 0; n < NTILE; ++n) {
    v8f acc = {};  // T tile = A(16x224) * M(:,16n..16n+15)
    const float* mp = Mw + (size_t)(n * KTILES) * 64 + 2 * lane;
#pragma unroll 4
    for (int kt = 0; kt < KTILES; ++kt) {
      // A-operand: lane holds A[r][4kt + 2*hi + {0,1}]  (LDS b64)
      v2f aop = *(const v2f*)(As + r * ASTRIDE + 4 * kt + 2 * hi);
      // B-operand: pre-swizzled M  (global b64, fully coalesced, L2 resident)
      v2f bop = *(const v2f*)(mp);
      mp += 64;
      acc = __builtin_amdgcn_wmma_f32_16x16x4_f32(false, aop, false, bop,
                                                  (short)0, acc, false, false);
    }
    // relayout T tile (C/D layout) -> A-operand layout through LDS scratch
#pragma unroll
    for (int v = 0; v < 8; ++v) Ts[(v + 8 * hi) * TSTRIDE + r] = acc[v];
    __builtin_amdgcn_wave_barrier();
#pragma unroll
    for (int kk = 0; kk < 4; ++kk) {
      // A-operand = T tile slice (K=4)
      v2f aopT = *(const v2f*)(Ts + r * TSTRIDE + 4 * kk + 2 * hi);
      // B-operand representing A^T: identical per-lane data to A-operand of A
      v2f bop2 = *(const v2f*)(As + r * ASTRIDE + 16 * n + 4 * kk + 2 * hi);
      acc2 = __builtin_amdgcn_wmma_f32_16x16x4_f32(false, aopT, false, bop2,
                                                   (short)0, acc2, false, false);
    }
    __builtin_amdgcn_wave_barrier();  // Ts reused next iteration (in-order DS)
  }

  // ---- extract diagonal of acc2: row (v + 8*hi) == col (lane&15) ----
#pragma unroll
  for (int v = 0; v < 8; ++v)
    if (v + 8 * hi == r) Ts[r] = acc2[v];
  __builtin_amdgcn_wave_barrier();

  if (lane < 16 && base + lane < nB) {
    float z = lin + Ts[lane];
    out[base + lane] = 1.f / (1.f + __expf(-z));
  }
}

// ---------------------------------------------------------------------------
extern "C" void kernel_launch(void* const* d_in, const int* in_sizes, int n_in,
                              void* d_out, int out_size, void* d_ws, size_t ws_size,
                              hipStream_t stream) {
  const float* x_dense = (const float*)d_in[0];
  const int*   x_sparse = (const int*)d_in[1];
  const float* emb = (const float*)d_in[2];
  const float* fm = (const float*)d_in[3];
  const float* W = (const float*)d_in[4];
  const float* bl = (const float*)d_in[5];
  float* Mw = (float*)d_ws;  // needs 224*224*4 = 200704 bytes of scratch
  float* out = (float*)d_out;
  int nB = out_size;

  ffm_build_M<<<(FEATP * FEATP + 255) / 256, 256, 0, stream>>>(fm, Mw);
  ffm_main<<<(nB + 63) / 64, 128, 0, stream>>>(x_dense, x_sparse, emb, W, bl,
                                               Mw, out, nB);
}